// RoIPlus_31327491457501
// MI455X (gfx1250) — compile-verified
//
#include <hip/hip_runtime.h>

// RoI Align (14x14, spatial_scale = 1/16) for MI455X / gfx1250.
// Memory-bound gather: features (2,512,100,152) f32 fit in the 192MB L2;
// the 205MB output is streamed with non-temporal stores so it never evicts
// the feature map. Bin geometry (tap offsets + bilinear weights) is computed
// once per (roi,bin,channel slice) in registers and reused across its 128 channels.

namespace {
constexpr int N_  = 2;
constexpr int C_  = 512;
constexpr int H_  = 100;
constexpr int W_  = 152;
constexpr int R_  = 512;
constexpr int AH_ = 14;
constexpr int AW_ = 14;
constexpr int NBIN = AH_ * AW_;        // 196 bins per (roi, channel)
constexpr int HW   = H_ * W_;          // 15200 elements per channel plane
constexpr float SCALE = 0.0625f;
constexpr int CSPLIT = 4;              // channel slices per roi
constexpr int CCHUNK = C_ / CSPLIT;    // 128 channels per block
}

__global__ __launch_bounds__(224) void roi_align_gather(
    const float* __restrict__ features,
    const float* __restrict__ rois,
    float* __restrict__ out)
{
    const int r = blockIdx.x;                 // wave-uniform -> scalar loads
    const int c_begin = blockIdx.y * CCHUNK;

    // ---- per-ROI parameters (uniform, SMEM path) ----
    const float bidx = rois[r * 5 + 0];
    const float xs0  = rois[r * 5 + 1] * SCALE;
    const float ys0  = rois[r * 5 + 2] * SCALE;
    const float xs1  = rois[r * 5 + 3] * SCALE;
    const float ys1  = rois[r * 5 + 4] * SCALE;
    const int   b    = (int)bidx;
    const float bin_w = fmaxf(xs1 - xs0, 0.0f) * (1.0f / (float)(AW_ - 1));
    const float bin_h = fmaxf(ys1 - ys0, 0.0f) * (1.0f / (float)(AH_ - 1));

    // ---- per-bin geometry, computed once, lives in registers ----
    const int bin = threadIdx.x;
    if (bin >= NBIN) return;                  // lanes 196..223 idle

    const int ah = bin / AW_;
    const int aw = bin - ah * AW_;

    float yf = fminf(fmaxf(ys0 + (float)ah * bin_h, 0.0f), (float)(H_ - 1));
    float xf = fminf(fmaxf(xs0 + (float)aw * bin_w, 0.0f), (float)(W_ - 1));
    const float y0f = floorf(yf);
    const float x0f = floorf(xf);
    const int y0 = (int)y0f;
    const int x0 = (int)x0f;
    const int y1 = min(y0 + 1, H_ - 1);
    const int x1 = min(x0 + 1, W_ - 1);
    const float wy = yf - y0f;
    const float wx = xf - x0f;

    // bilinear weights exactly as the reference combines them
    const float w00 = (1.0f - wy) * (1.0f - wx);
    const float w01 = (1.0f - wy) * wx;
    const float w10 = wy * (1.0f - wx);
    const float w11 = wy * wx;

    // tap offsets within one channel plane (per-lane constant voffsets)
    const int off00 = y0 * W_ + x0;
    const int off01 = y0 * W_ + x1;
    const int off10 = y1 * W_ + x0;
    const int off11 = y1 * W_ + x1;

    const float* __restrict__ fb =
        features + (size_t)b * (size_t)(C_ * HW) + (size_t)c_begin * HW;
    float* __restrict__ ob =
        out + (size_t)r * (size_t)(C_ * NBIN) + (size_t)c_begin * NBIN + bin;

    // ---- channel loop: 4 loads + 4 FMA + 1 NT store per output ----
    for (int c = 0; c < CCHUNK; c += 4) {
        const float* __restrict__ p = fb + c * HW;
        if (c + 4 < CCHUNK) {
            // pull next channel group toward the WGP (global_prefetch_b8)
            __builtin_prefetch((const void*)(p + 4 * HW + off00), 0, 3);
        }
#pragma unroll
        for (int u = 0; u < 4; ++u) {
            const float* __restrict__ q = p + u * HW;
            const float v00 = q[off00];
            const float v01 = q[off01];
            const float v10 = q[off10];
            const float v11 = q[off11];
            float val = v00 * w00;
            val = __builtin_fmaf(v01, w01, val);
            val = __builtin_fmaf(v10, w10, val);
            val = __builtin_fmaf(v11, w11, val);
            // streaming store: keep the 62MB feature map resident in L2,
            // don't let the 205MB output evict it (CDNA5 TH=NT policy)
            __builtin_nontemporal_store(val, ob + (c + u) * NBIN);
        }
    }
}

extern "C" void kernel_launch(void* const* d_in, const int* in_sizes, int n_in,
                              void* d_out, int out_size, void* d_ws, size_t ws_size,
                              hipStream_t stream) {
    const float* features = (const float*)d_in[0];
    const float* rois     = (const float*)d_in[1];
    float*       out      = (float*)d_out;

    dim3 grid(R_, CSPLIT);   // 512 rois x 4 channel slices = 2048 blocks
    dim3 block(224);         // 7 wave32; 196 active bin lanes
    roi_align_gather<<<grid, block, 0, stream>>>(features, rois, out);
}